// Aij_29575144800551
// MI455X (gfx1250) — compile-verified
//
#include <hip/hip_runtime.h>

typedef __attribute__((ext_vector_type(2))) float v2f;
typedef __attribute__((ext_vector_type(8))) float v8f;

#define BATCH 8
#define NNODE 2048
#define FDIM  64
#define ROWS  (BATCH * NNODE)   // 16384 total (b,n) rows

// ---------------------------------------------------------------------------
// Kernel A: fold attention vectors through the projection matrix.
//   w_s_eff[i] = sum_o kernel[i,0,o] * attn_self[o]
//   w_n_eff[i] = sum_o kernel[i,0,o] * attn_neigh[o]
// 64 threads, one per input feature i. Trivial cost.
// ---------------------------------------------------------------------------
__global__ void gat_eff_weights(const float* __restrict__ Wk,   // [F][1][F]
                                const float* __restrict__ As,   // [F]
                                const float* __restrict__ An,   // [F]
                                float* __restrict__ wse,
                                float* __restrict__ wne) {
    const int i = threadIdx.x;            // 0..63
    float s = 0.f, n = 0.f;
#pragma unroll
    for (int o = 0; o < FDIM; ++o) {
        const float k = Wk[i * FDIM + o];
        s = fmaf(k, As[o], s);
        n = fmaf(k, An[o], n);
    }
    wse[i] = s;
    wne[i] = n;
}

// ---------------------------------------------------------------------------
// Kernel B: score GEMM via V_WMMA_F32_16X16X4_F32 (wave32).
//   D[16x16] = A[16x4] * B[4x16] accumulated over K=64 (16 WMMA steps).
//   A = 16 node rows of encode; B col 0 = w_s_eff, col 1 = w_n_eff, rest 0.
//   D column 0 -> S[row], column 1 -> T[row].
// Layouts (ISA 7.12.2, 32-bit):
//   A: lane l: M = l%16; VGPR0/1 hold K = 2*(l/16) + {0,1}
//   B: lane l: N = l%16; VGPR0/1 hold K = 2*(l/16) + {0,1}
//   D: lane l: N = l%16; VGPR r holds M = r + 8*(l/16)
// B-fragment gather is branch-free: LDS holds a 3-column table
// {self, neigh, zeros}; each lane picks its column base ONCE before the
// loop, so the inner loop is ds_load_b64 + global_load_b64 + v_wmma only
// (uniform control flow, EXEC stays all-1s as WMMA requires).
// ---------------------------------------------------------------------------
__global__ void gat_scores_wmma(const float* __restrict__ E,   // [ROWS][F]
                                const float* __restrict__ wse,
                                const float* __restrict__ wne,
                                float* __restrict__ S,
                                float* __restrict__ T) {
    __shared__ float wl[3 * FDIM];        // [0:64)=self, [64:128)=neigh, [128:192)=0
    const int tid = threadIdx.x;
    if (tid < 3 * FDIM)
        wl[tid] = (tid < FDIM) ? wse[tid]
                : (tid < 2 * FDIM) ? wne[tid - FDIM]
                : 0.f;
    __syncthreads();

    const int lane  = tid & 31;
    const int wave  = tid >> 5;
    const int row0  = (blockIdx.x * (blockDim.x >> 5) + wave) * 16;
    const int mn    = lane & 15;           // A: row M, B/D: col N
    const int khalf = (lane >> 4) << 1;    // K sub-offset per lane half

    const float* arow = E + (size_t)(row0 + mn) * FDIM + khalf;
    // loop-invariant per-lane column base (no divergence in the loop)
    const float* wcol = wl + ((mn == 0) ? 0 : ((mn == 1) ? FDIM : 2 * FDIM)) + khalf;

    v8f acc = {};
#pragma unroll
    for (int k0 = 0; k0 < FDIM; k0 += 4) {
        const v2f a = *(const v2f*)(arow + k0);   // global_load_b64 (8B aligned)
        const v2f b = *(const v2f*)(wcol + k0);   // ds_load_b64     (8B aligned)
        acc = __builtin_amdgcn_wmma_f32_16x16x4_f32(
            /*neg_a=*/false, a, /*neg_b=*/false, b,
            /*c_mod=*/(short)0, acc, /*reuse_a=*/false, /*reuse_b=*/false);
    }

    if (mn < 2) {
        float* dst = (mn == 0) ? S : T;
        const int base = row0 + 8 * (lane >> 4);
#pragma unroll
        for (int r = 0; r < 8; ++r) dst[base + r] = acc[r];
    }
}

// ---------------------------------------------------------------------------
// Kernel C: out[b,n,m] = softmax_m( leaky_relu(s[b,n] + t[b,m], 0.2) )
// One row per wave, 8 waves (= 8 rows) per block; t[b,:] staged in LDS once
// per block. Logits cached in 16 float4 regs per lane -> single exp and single
// 128-bit coalesced store per output element (bandwidth-floor bound).
// ---------------------------------------------------------------------------
__device__ __forceinline__ float lrelu(float x) { return x > 0.f ? x : 0.2f * x; }

__global__ void gat_softmax(const float* __restrict__ S,
                            const float* __restrict__ T,
                            float* __restrict__ out) {
    __shared__ float4 tl4[NNODE / 4];          // 8 KB: t-row for this batch

    const int tid  = threadIdx.x;
    const int lane = tid & 31;
    const int wave = tid >> 5;                 // 0..7
    const int b    = blockIdx.x >> 8;          // 256 blocks per batch
    const int row  = ((blockIdx.x & 255) << 3) + wave;

    // cooperative stage of t[b, 0..2047] into LDS (b128 loads/stores)
    const float4* tb4 = (const float4*)(T + (size_t)b * NNODE);
    for (int i = tid; i < NNODE / 4; i += blockDim.x) tl4[i] = tb4[i];
    __syncthreads();

    const float sv = S[(size_t)b * NNODE + row];

    // Pass 1: logits (leaky-relu) into registers + running max
    float4 e[16];
    float mx = -3.402823466e38f;
#pragma unroll
    for (int it = 0; it < 16; ++it) {
        const float4 v = tl4[lane + 32 * it];
        float4 x;
        x.x = lrelu(sv + v.x);
        x.y = lrelu(sv + v.y);
        x.z = lrelu(sv + v.z);
        x.w = lrelu(sv + v.w);
        mx = fmaxf(mx, fmaxf(fmaxf(x.x, x.y), fmaxf(x.z, x.w)));
        e[it] = x;
    }
#pragma unroll
    for (int off = 16; off > 0; off >>= 1)
        mx = fmaxf(mx, __shfl_xor(mx, off, 32));

    // Pass 2: exponentiate in-register, wave-reduce the sum
    float sum = 0.f;
#pragma unroll
    for (int it = 0; it < 16; ++it) {
        float4 x = e[it];
        x.x = __expf(x.x - mx);
        x.y = __expf(x.y - mx);
        x.z = __expf(x.z - mx);
        x.w = __expf(x.w - mx);
        sum += (x.x + x.y) + (x.z + x.w);
        e[it] = x;
    }
#pragma unroll
    for (int off = 16; off > 0; off >>= 1)
        sum += __shfl_xor(sum, off, 32);
    const float inv = 1.0f / sum;

    // Pass 3: scale + coalesced 128-bit stores (H=1 => mean over heads is id)
    float4* o4 = (float4*)(out + ((size_t)b * NNODE + row) * NNODE);
#pragma unroll
    for (int it = 0; it < 16; ++it) {
        float4 x = e[it];
        x.x *= inv; x.y *= inv; x.z *= inv; x.w *= inv;
        o4[lane + 32 * it] = x;
    }
}

// ---------------------------------------------------------------------------
extern "C" void kernel_launch(void* const* d_in, const int* in_sizes, int n_in,
                              void* d_out, int out_size, void* d_ws, size_t ws_size,
                              hipStream_t stream) {
    const float* encode = (const float*)d_in[0];   // [8,2048,64]
    const float* Wk     = (const float*)d_in[1];   // [64,1,64]
    const float* As     = (const float*)d_in[2];   // [64,1,1]
    const float* An     = (const float*)d_in[3];   // [64,1,1]
    float*       out    = (float*)d_out;           // [8,2048,2048]

    float* ws  = (float*)d_ws;
    float* wse = ws;                 // 64 floats
    float* wne = ws + FDIM;          // 64 floats
    float* S   = ws + 2 * FDIM;      // 16384 floats
    float* T   = S + ROWS;           // 16384 floats

    gat_eff_weights<<<1, FDIM, 0, stream>>>(Wk, As, An, wse, wne);

    // 16384 rows / (16 rows per wave * 8 waves per block) = 128 blocks
    gat_scores_wmma<<<ROWS / (16 * 8), 256, 0, stream>>>(encode, wse, wne, S, T);

    // 8 rows per block -> 2048 blocks
    gat_softmax<<<BATCH * (NNODE / 8), 256, 0, stream>>>(S, T, out);
}